// CliffordFourierBasicBlock3d_73005854098019
// MI455X (gfx1250) — compile-verified
//
#include <hip/hip_runtime.h>
#include <math.h>

typedef __attribute__((ext_vector_type(2))) float v2f;
typedef __attribute__((ext_vector_type(8))) float v8f;

#define SPAT 262144           // 64^3
#define TWO_PI 6.28318530717958647692f

// Clifford table M[p,q,k]: for each (k,q) exactly one p contributes with sign s.
__constant__ int c_P[64] = {
  0,1,2,3,4,5,6,7,
  1,0,4,5,2,3,7,6,
  2,4,0,6,1,7,3,5,
  3,5,6,0,7,1,2,4,
  4,2,1,7,0,6,5,3,
  5,3,7,1,6,0,4,2,
  6,7,3,2,5,4,0,1,
  7,6,5,4,3,2,1,0};
__constant__ float c_S[64] = {
  1, 1, 1, 1,-1,-1,-1,-1,
  1, 1, 1, 1,-1,-1,-1,-1,
  1,-1, 1, 1, 1, 1,-1, 1,
  1,-1,-1, 1,-1, 1, 1,-1,
  1,-1, 1, 1, 1, 1,-1, 1,
  1,-1,-1, 1,-1, 1, 1,-1,
  1, 1,-1, 1, 1,-1, 1, 1,
  1, 1,-1, 1, 1,-1, 1, 1};

__device__ __forceinline__ float gelu_exact(float v) {
  return 0.5f * v * (1.0f + erff(v * 0.70710678118654752440f));
}

// ---- CDNA5 async memory->LDS staging (ASYNCcnt path) ----------------------
typedef __attribute__((address_space(3))) void lds_void;

__device__ __forceinline__ void async_g2l_b128(void* lds, const void* g) {
  lds_void* lp = (lds_void*)lds;          // addrspace(3): 32-bit LDS offset
  asm volatile("global_load_async_to_lds_b128 %0, %1, off"
               :: "v"(lp), "v"(g) : "memory");
}
__device__ __forceinline__ void async_g2l_b64(void* lds, const void* g) {
  lds_void* lp = (lds_void*)lds;
  asm volatile("global_load_async_to_lds_b64 %0, %1, off"
               :: "v"(lp), "v"(g) : "memory");
}
__device__ __forceinline__ void wait_async0() {
  asm volatile("s_wait_asynccnt 0x0" ::: "memory");
}

// ---------------------------------------------------------------------------
// k_init: build Wk[128][128] (combined Clifford conv weights), forward DFT
// table F[16f][64n] = exp(-2pi i (f-8) n / 64), inverse table
// G[64n][16f] = exp(+2pi i (f-8) n / 64) / 64  (1/64 per inverse axis).
// ---------------------------------------------------------------------------
__global__ __launch_bounds__(256) void k_init(const float* __restrict__ w_conv,
                                              float* __restrict__ Wk,
                                              float* __restrict__ Ftab,
                                              float* __restrict__ Gtab) {
  const int t = threadIdx.x;
  for (int idx = t; idx < 16384; idx += 256) {
    int r = idx >> 7, cc = idx & 127;
    int o = r >> 3, k = r & 7, i = cc >> 3, q = cc & 7;
    Wk[idx] = c_S[k*8+q] * w_conv[c_P[k*8+q]*256 + o*16 + i];   // w_conv (8p,16o,16i)
  }
  for (int idx = t; idx < 1024; idx += 256) {
    int f = idx >> 6, z = idx & 63;
    float ang = -(TWO_PI / 64.0f) * (float)((f - 8) * z);
    Ftab[2*idx]   = cosf(ang);
    Ftab[2*idx+1] = sinf(ang);
  }
  for (int idx = t; idx < 1024; idx += 256) {
    int xx = idx >> 4, f = idx & 15;
    float ang = (TWO_PI / 64.0f) * (float)((f - 8) * xx);
    Gtab[2*idx]   = cosf(ang) * (1.0f/64.0f);
    Gtab[2*idx+1] = sinf(ang) * (1.0f/64.0f);
  }
}

// ---------------------------------------------------------------------------
// x2 path: out[(o,s,k)] = sum_{i,q} Wk[(o*8+k)][(i*8+q)] * x[(i,s,q)]
// 128 x 128 x 262144 GEMM via V_WMMA_F32_16X16X4_F32.
// ---------------------------------------------------------------------------
__global__ __launch_bounds__(256) void k_conv_wmma(const float* __restrict__ x,
                                                   const float* __restrict__ Wk,
                                                   float* __restrict__ out) {
  const int tid  = threadIdx.x;
  const int wave = tid >> 5;
  const int lane = tid & 31;
  const int lm   = lane & 15;   // M row (A) / N col (B,C)
  const int lh   = lane >> 4;   // K-half selector

  v2f a[32];
#pragma unroll
  for (int c = 0; c < 32; ++c)
    a[c] = *reinterpret_cast<const v2f*>(Wk + (wave*16 + lm)*128 + c*4 + lh*2);

  for (int ti = 0; ti < 16; ++ti) {
    const int tile = blockIdx.x * 16 + ti;
    const int s0 = tile * 16;
    const int s  = s0 + lm;
    // prefetch next tile's B slabs (global_prefetch_b8); lanes 0..15 cover i=0..15
    __builtin_prefetch(x + ((size_t)lm * SPAT + s0 + 16) * 8, 0, 1);
    v8f c = {};
#pragma unroll
    for (int ch = 0; ch < 32; ++ch) {
      const int i  = ch >> 1;
      const int q0 = (ch & 1) * 4 + lh * 2;          // K = ch*4 + lh*2 + {0,1}
      const v2f b = *reinterpret_cast<const v2f*>(x + ((size_t)i * SPAT + s) * 8 + q0);
      c = __builtin_amdgcn_wmma_f32_16x16x4_f32(false, a[ch], false, b,
                                                (short)0, c, false, false);
    }
#pragma unroll
    for (int j = 0; j < 8; ++j) {
      const int r = wave*16 + j + lh*8;
      const int o = r >> 3, k = r & 7;
      out[((size_t)o * SPAT + s0 + lm) * 8 + k] = c[j];
    }
  }
}

// ---------------------------------------------------------------------------
// Forward cropped-DFT stage Z: out1[i,x,y,f,q] = sum_z x[i,x,y,z,q] F[f][z]
// LDS staged via async memory->LDS (no VGPR round trip).
// ---------------------------------------------------------------------------
__global__ __launch_bounds__(128) void k_fwd_z(const float* __restrict__ x,
                                               const float* __restrict__ Ftab,
                                               float* __restrict__ out1) {
  __shared__ __align__(16) float  shx[512];
  __shared__ __align__(16) float2 shF[1024];
  const int b = blockIdx.x;               // ((i*64+x)*64+y)
  const int t = threadIdx.x;              // f*8+q
  const float* src = x + (size_t)b * 512;
  async_g2l_b128(&shx[t*4], src + t*4);                 // 512 floats
#pragma unroll
  for (int j = 0; j < 4; ++j) {
    const int e = t + j*128;                            // 512 x b128 = 2048 floats
    async_g2l_b128(&shF[e*2], Ftab + e*4);
  }
  wait_async0();
  __syncthreads();
  const int f = t >> 3, q = t & 7;
  float re = 0.f, im = 0.f;
#pragma unroll 8
  for (int z = 0; z < 64; ++z) {
    float  v = shx[z*8 + q];
    float2 w = shF[f*64 + z];
    re += v * w.x; im += v * w.y;
  }
  ((float2*)out1)[(size_t)b * 128 + t] = make_float2(re, im);
}

// Stage Y: out2[i,x,fy,fz,q] = sum_y out1[i,x,y,fz,q] F[fy][y]
__global__ __launch_bounds__(256) void k_fwd_y(const float* __restrict__ out1,
                                               const float* __restrict__ Ftab,
                                               float* __restrict__ out2) {
  __shared__ __align__(16) float2 sh[1024];             // [y][fz]
  const int b = blockIdx.x;               // i*512 + x*8 + q
  const int i = b >> 9, xx = (b >> 3) & 63, q = b & 7;
  const int t = threadIdx.x;
  const float2* in = (const float2*)out1;
  const size_t base = ((size_t)i*64 + xx) * 8192;
#pragma unroll
  for (int j = t; j < 1024; j += 256) {
    int y = j >> 4, fz = j & 15;
    async_g2l_b64(&sh[j], &in[base + (size_t)y*128 + fz*8 + q]);
  }
  wait_async0();
  __syncthreads();
  const float2* Fg = (const float2*)Ftab;
  const int fy = t >> 4, fz = t & 15;
  float2 acc = make_float2(0.f, 0.f);
#pragma unroll 8
  for (int y = 0; y < 64; ++y) {
    float2 v = sh[y*16 + fz];
    float2 w = Fg[fy*64 + y];
    acc.x += v.x*w.x - v.y*w.y;
    acc.y += v.x*w.y + v.y*w.x;
  }
  ((float2*)out2)[(size_t)i*131072 + (size_t)xx*2048 + fy*128 + fz*8 + q] = acc;
}

// Stage X: xc[i,fx,fy,fz,q] = sum_x out2[i,x,fy,fz,q] F[fx][x]
__global__ __launch_bounds__(256) void k_fwd_x(const float* __restrict__ out2,
                                               const float* __restrict__ Ftab,
                                               float* __restrict__ xc) {
  __shared__ __align__(16) float2 sh[1024];             // [x][fz]
  const int b = blockIdx.x;               // i*128 + fy*8 + q
  const int i = b >> 7, fy = (b >> 3) & 15, q = b & 7;
  const int t = threadIdx.x;
  const float2* in = (const float2*)out2;
#pragma unroll
  for (int j = t; j < 1024; j += 256) {
    int xx = j >> 4, fz = j & 15;
    async_g2l_b64(&sh[j], &in[(size_t)i*131072 + (size_t)xx*2048 + fy*128 + fz*8 + q]);
  }
  wait_async0();
  __syncthreads();
  const float2* Fg = (const float2*)Ftab;
  const int fx = t >> 4, fz = t & 15;
  float2 acc = make_float2(0.f, 0.f);
#pragma unroll 8
  for (int xx = 0; xx < 64; ++xx) {
    float2 v = sh[xx*16 + fz];
    float2 w = Fg[fx*64 + xx];
    acc.x += v.x*w.x - v.y*w.y;
    acc.y += v.x*w.y + v.y*w.x;
  }
  ((float2*)xc)[(size_t)i*32768 + fx*2048 + fy*128 + fz*8 + q] = acc;
}

// ---------------------------------------------------------------------------
// Spectral Clifford mix: oc[o,f,k] = sum_{q,i} S[k][q] w_spec[P[k][q],i,o,f] xc[i,f,q]
// ---------------------------------------------------------------------------
__global__ __launch_bounds__(128) void k_specmix(const float* __restrict__ xc,
                                                 const float* __restrict__ w_spec,
                                                 float* __restrict__ oc) {
  __shared__ float  shw[2048];            // [p][i][o]
  __shared__ float2 shx[128];             // [i][q]
  const int foff = blockIdx.x;            // fx*256+fy*16+fz
  const int t = threadIdx.x;
  for (int j = t; j < 2048; j += 128) shw[j] = w_spec[(size_t)j * 4096 + foff];
  {
    const int i = t >> 3, q = t & 7;
    shx[t] = ((const float2*)xc)[(size_t)i*32768 + (size_t)foff*8 + q];
  }
  __syncthreads();
  const int o = t >> 3, k = t & 7;
  float2 acc = make_float2(0.f, 0.f);
#pragma unroll
  for (int q = 0; q < 8; ++q) {
    const int   p  = c_P[k*8+q];
    const float sg = c_S[k*8+q];
    const float* wrow = shw + p*256 + o;
#pragma unroll
    for (int i = 0; i < 16; ++i) {
      float  wv = sg * wrow[i*16];
      float2 xv = shx[i*8+q];
      acc.x += wv * xv.x;
      acc.y += wv * xv.y;
    }
  }
  ((float2*)oc)[(size_t)o*32768 + (size_t)foff*8 + k] = acc;
}

// ---------------------------------------------------------------------------
// Inverse stages (adjoint DFT, 1/64 folded into G). Inputs live in L2.
// ---------------------------------------------------------------------------
__global__ __launch_bounds__(256) void k_inv_z(const float* __restrict__ oc,
                                               const float* __restrict__ Gtab,
                                               float* __restrict__ inv1) {
  const int b = blockIdx.x;               // o*512 + ff*2 + half
  const int o = b >> 9, ff = (b >> 1) & 255, half = b & 1;
  const int e = half*256 + threadIdx.x;   // z*8+k
  const int z = e >> 3, k = e & 7;
  const float2* in = (const float2*)oc + (size_t)o*32768 + ff*128 + k;
  const float2* Gg = (const float2*)Gtab + z*16;
  float2 acc = make_float2(0.f, 0.f);
#pragma unroll
  for (int fz = 0; fz < 16; ++fz) {
    float2 v = in[fz*8];
    float2 w = Gg[fz];
    acc.x += v.x*w.x - v.y*w.y;
    acc.y += v.x*w.y + v.y*w.x;
  }
  ((float2*)inv1)[((size_t)o*256 + ff)*512 + e] = acc;
}

__global__ __launch_bounds__(256) void k_inv_y(const float* __restrict__ inv1,
                                               const float* __restrict__ Gtab,
                                               float* __restrict__ inv2) {
  const int b = blockIdx.x;               // ((o*16+fx)*64+y)*2 + half
  const int o = b >> 11, fx = (b >> 7) & 15, y = (b >> 1) & 63, half = b & 1;
  const int e = half*256 + threadIdx.x;   // z*8+k
  const float2* in = (const float2*)inv1 + ((size_t)o*256 + fx*16)*512 + e;
  const float2* Gg = (const float2*)Gtab + y*16;
  float2 acc = make_float2(0.f, 0.f);
#pragma unroll
  for (int fy = 0; fy < 16; ++fy) {
    float2 v = in[(size_t)fy*512];
    float2 w = Gg[fy];
    acc.x += v.x*w.x - v.y*w.y;
    acc.y += v.x*w.y + v.y*w.x;
  }
  ((float2*)inv2)[(((size_t)o*16 + fx)*64 + y)*512 + e] = acc;
}

// Final: x1 = Re(sum_fx inv2 * G[x][fx]); out = gelu(x1 + x2(out) + bias)
__global__ __launch_bounds__(256) void k_final(const float* __restrict__ inv2,
                                               const float* __restrict__ Gtab,
                                               const float* __restrict__ b_conv,
                                               float* __restrict__ out) {
  const int b = blockIdx.x;               // o*4096 + x*64 + y
  const int o = b >> 12, xx = (b >> 6) & 63, y = b & 63;
  const float2* Gg = (const float2*)Gtab + xx*16;
  const float2* in = (const float2*)inv2 + ((size_t)o*1024 + y)*512;
  float* op = out + (size_t)b * 512;
#pragma unroll
  for (int half = 0; half < 2; ++half) {
    const int e = half*256 + threadIdx.x; // z*8+k
    const int k = e & 7;
    float acc = 0.f;
#pragma unroll
    for (int fx = 0; fx < 16; ++fx) {
      float2 v = in[(size_t)fx*32768 + e];
      float2 w = Gg[fx];
      acc += v.x*w.x - v.y*w.y;           // real part only
    }
    float v = acc + op[e] + b_conv[o*8 + k];
    op[e] = gelu_exact(v);
  }
}

// ---------------------------------------------------------------------------
extern "C" void kernel_launch(void* const* d_in, const int* in_sizes, int n_in,
                              void* d_out, int out_size, void* d_ws, size_t ws_size,
                              hipStream_t stream) {
  const float* x      = (const float*)d_in[0];   // (1,16,64,64,64,8)
  const float* w_spec = (const float*)d_in[1];   // (8,16,16,16,16,16)
  const float* w_conv = (const float*)d_in[2];   // (8,16,16)
  const float* b_conv = (const float*)d_in[3];   // (16,8)
  float* out = (float*)d_out;                    // (1,16,64,64,64,8)

  float* ws   = (float*)d_ws;
  float* Wk   = ws;                 // 16384 f
  float* Ftab = ws + 16384;         //  2048 f (16x64 cplx)
  float* Gtab = ws + 18432;         //  2048 f (64x16 cplx, /64)
  float* bufA = ws + 20480;         // 16,777,216 f : out1 / inv2
  float* bufB = bufA + 16777216;    //  4,194,304 f : out2 / inv1
  float* bufC = bufB + 4194304;     //  1,048,576 f : xc
  float* bufD = bufC + 1048576;     //  1,048,576 f : oc   (total ~92.4 MB)

  k_init     <<<1,     256, 0, stream>>>(w_conv, Wk, Ftab, Gtab);
  k_conv_wmma<<<1024,  256, 0, stream>>>(x, Wk, out);
  k_fwd_z    <<<65536, 128, 0, stream>>>(x, Ftab, bufA);
  k_fwd_y    <<<8192,  256, 0, stream>>>(bufA, Ftab, bufB);
  k_fwd_x    <<<2048,  256, 0, stream>>>(bufB, Ftab, bufC);
  k_specmix  <<<4096,  128, 0, stream>>>(bufC, w_spec, bufD);
  k_inv_z    <<<8192,  256, 0, stream>>>(bufD, Gtab, bufB);
  k_inv_y    <<<32768, 256, 0, stream>>>(bufB, Gtab, bufA);
  k_final    <<<65536, 256, 0, stream>>>(bufA, Gtab, b_conv, out);

  (void)in_sizes; (void)n_in; (void)out_size; (void)ws_size;
}